// Contrast_84456236908569
// MI455X (gfx1250) — compile-verified
//
#include <hip/hip_runtime.h>
#include <math.h>

#define N_  4096
#define D_  256
#define K0_ 64
#define K1_ 128
#define TAU_ 0.8f
#define LAM_ 0.5f

typedef __attribute__((ext_vector_type(2))) float v2f;
typedef __attribute__((ext_vector_type(8))) float v8f;

__device__ __forceinline__ float waveReduceSum(float v) {
    v += __shfl_xor(v, 1, 32);
    v += __shfl_xor(v, 2, 32);
    v += __shfl_xor(v, 4, 32);
    v += __shfl_xor(v, 8, 32);
    v += __shfl_xor(v, 16, 32);
    return v;
}
// reduce within each 16-lane half (keeps WMMA C-layout halves separate)
__device__ __forceinline__ float halfReduceSum(float v) {
    v += __shfl_xor(v, 1, 32);
    v += __shfl_xor(v, 2, 32);
    v += __shfl_xor(v, 4, 32);
    v += __shfl_xor(v, 8, 32);
    return v;
}

// OUT = act(X @ W^T + b). X: N x D, W: D x D (torch Linear). One wave per 16x16 tile.
__global__ void __launch_bounds__(128)
gemm_xwt_bias(const float* __restrict__ X, const float* __restrict__ W,
              const float* __restrict__ bias, float* __restrict__ OUT, int actElu) {
    const int lane = threadIdx.x & 31;
    const int wave = threadIdx.x >> 5;
    const int tile = blockIdx.x * 4 + wave;
    const int NT = D_ / 16;
    const int tm = tile / NT;
    const int tn = tile % NT;
    if (tm >= N_ / 16) return;

    const int m    = lane & 15;
    const int half = lane >> 4;

    const float* Arow = X + (size_t)(tm * 16 + m) * D_; // A: M = lane%16
    const float* Brow = W + (size_t)(tn * 16 + m) * D_; // B col n = lane%16; B[k,n]=W[n,k]

    v8f acc = {};
    #pragma unroll 8
    for (int k = 0; k < D_; k += 4) {
        const int kb = k + half * 2; // lanes<16 hold K,K+1; lanes>=16 hold K+2,K+3
        v2f a = *(const v2f*)(Arow + kb);
        v2f b = *(const v2f*)(Brow + kb);
        acc = __builtin_amdgcn_wmma_f32_16x16x4_f32(false, a, false, b,
                                                    (short)0, acc, false, false);
    }
    const float bv = bias[tn * 16 + m];
    #pragma unroll
    for (int v = 0; v < 8; ++v) {
        const int orow = tm * 16 + v + half * 8;
        float val = acc[v] + bv;
        if (actElu) val = val > 0.f ? val : (__expf(val) - 1.f);
        OUT[(size_t)orow * D_ + tn * 16 + m] = val;
    }
}

__global__ void __launch_bounds__(256)
row_norms(const float* __restrict__ Z, float* __restrict__ nrm) {
    const int lane = threadIdx.x & 31;
    const int row = blockIdx.x * 8 + (threadIdx.x >> 5);
    if (row >= N_) return;
    const float* p = Z + (size_t)row * D_;
    float s = 0.f;
    #pragma unroll
    for (int d = lane; d < D_; d += 32) { float x = p[d]; s += x * x; }
    s = waveReduceSum(s);
    if (lane == 0) nrm[row] = sqrtf(s);
}

// Fused sim+NCE pass: for each row i of exp(cos(ZA_i, ZB_j)/tau) compute
// rowsum[i] and rowpos[i] = sum_j s*pos[i,j] without materializing the NxN matrix.
__global__ void __launch_bounds__(64)
sim_nce_pass(const float* __restrict__ ZA, const float* __restrict__ ZB,
             const float* __restrict__ nA, const float* __restrict__ nB,
             const float* __restrict__ pos, float invTau,
             float* __restrict__ rowsum, float* __restrict__ rowpos) {
    __shared__ float ldsA[2][16 * 258]; // stride 258: conflict-free, 8B-aligned rows
    const int lane = threadIdx.x & 31;
    const int wave = threadIdx.x >> 5;
    const int itile = blockIdx.x * 2 + wave;
    const int i0 = itile * 16;
    float* la = ldsA[wave];

    // stage this wave's 16 x 256 A tile into LDS
    for (int idx = lane; idx < 16 * D_; idx += 32) {
        const int r = idx >> 8, c = idx & (D_ - 1);
        la[r * 258 + c] = ZA[(size_t)(i0 + r) * D_ + c];
    }

    const int m    = lane & 15;
    const int half = lane >> 4;
    float invA[8];
    #pragma unroll
    for (int v = 0; v < 8; ++v) invA[v] = 1.f / nA[i0 + v + half * 8];

    float rs[8] = {0,0,0,0,0,0,0,0};
    float rp[8] = {0,0,0,0,0,0,0,0};

    for (int jt = 0; jt < N_ / 16; ++jt) {
        const int j0 = jt * 16;
        const float* Brow = ZB + (size_t)(j0 + m) * D_;
        const float* Alds = la + m * 258;
        v8f acc = {};
        #pragma unroll 8
        for (int k = 0; k < D_; k += 4) {
            const int kb = k + half * 2;
            v2f a = *(const v2f*)(Alds + kb);
            v2f b = *(const v2f*)(Brow + kb);
            acc = __builtin_amdgcn_wmma_f32_16x16x4_f32(false, a, false, b,
                                                        (short)0, acc, false, false);
        }
        const float invB = invTau / nB[j0 + m];
        #pragma unroll
        for (int v = 0; v < 8; ++v) {
            const int gi = i0 + v + half * 8;
            const int gj = j0 + m;
            const float s = __expf(acc[v] * invA[v] * invB);
            const float pv = pos[(size_t)gi * N_ + gj];
            rs[v] += s;
            rp[v] += s * pv;
        }
    }
    #pragma unroll
    for (int v = 0; v < 8; ++v) { rs[v] = halfReduceSum(rs[v]); rp[v] = halfReduceSum(rp[v]); }
    if ((lane & 15) == 0) {
        #pragma unroll
        for (int v = 0; v < 8; ++v) {
            const int gi = i0 + v + half * 8;
            rowsum[gi] = rs[v];
            rowpos[gi] = rp[v];
        }
    }
}

// loss_slot = mean_i [ log(rowsum+1e-8) - log(rowpos) ]  (single block)
__global__ void __launch_bounds__(256)
nce_finalize(const float* __restrict__ rowsum, const float* __restrict__ rowpos,
             float* __restrict__ slot) {
    __shared__ float red[256];
    float s = 0.f;
    for (int i = threadIdx.x; i < N_; i += 256)
        s += logf(rowsum[i] + 1e-8f) - logf(rowpos[i]);
    red[threadIdx.x] = s;
    __syncthreads();
    for (int off = 128; off > 0; off >>= 1) {
        if (threadIdx.x < off) red[threadIdx.x] += red[threadIdx.x + off];
        __syncthreads();
    }
    if (threadIdx.x == 0) *slot = red[0] / (float)N_;
}

// weight@z collapses to cluster/pair scatter sums (weight built from equality masks)
__global__ void __launch_bounds__(256)
pocl_scatter(const float* __restrict__ Z, const int* __restrict__ c0,
             const int* __restrict__ c1, float* __restrict__ S0,
             float* __restrict__ SP, float* __restrict__ n0, float* __restrict__ npr) {
    const int idx = blockIdx.x * 256 + threadIdx.x;
    if (idx >= N_ * D_) return;
    const int a = idx >> 8;
    const int d = idx & (D_ - 1);
    const int k0 = c0[a], k1 = c1[a];
    const float z = Z[idx];
    atomicAdd(&S0[(size_t)k0 * D_ + d], z);
    atomicAdd(&SP[((size_t)k0 * K1_ + k1) * D_ + d], z);
    if (d == 0) { atomicAdd(&n0[k0], 1.f); atomicAdd(&npr[k0 * K1_ + k1], 1.f); }
}

__global__ void __launch_bounds__(256)
pocl_loss(const float* __restrict__ Zanc, const int* __restrict__ c0,
          const int* __restrict__ c1, const float* __restrict__ S0,
          const float* __restrict__ SP, const float* __restrict__ n0,
          const float* __restrict__ npr, const float* __restrict__ cent0,
          const float* __restrict__ cent1, const float* __restrict__ dens0,
          const float* __restrict__ dens1, float* __restrict__ slot) {
    const int lane = threadIdx.x & 31;
    const int a = blockIdx.x * 8 + (threadIdx.x >> 5);
    if (a >= N_) return;
    const int k0 = c0[a], k1 = c1[a];
    const int pr = k0 * K1_ + k1;
    const float cnt0 = n0[k0] + npr[pr]; // w0 rowsum
    const float cntp = npr[pr];          // pair count (factor 2 cancels)

    float zs[8];
    #pragma unroll
    for (int i = 0; i < 8; ++i) zs[i] = Zanc[(size_t)a * D_ + lane + 32 * i];

    float dot0 = 0.f, dot1 = 0.f;
    #pragma unroll
    for (int i = 0; i < 8; ++i) {
        const int d = lane + 32 * i;
        const float sp = SP[(size_t)pr * D_ + d];
        const float p0 = 0.1f * (S0[(size_t)k0 * D_ + d] + sp) / cnt0; // 0.1*c0
        const float p1 = p0 + 0.1f * sp / cntp;                       // 0.1*(c0+c1)
        dot0 += zs[i] * p0;
        dot1 += zs[i] * p1;
    }
    dot0 = waveReduceSum(dot0);
    dot1 = waveReduceSum(dot1);

    float neg0 = 0.f;
    for (int k = 0; k < K0_; ++k) {
        float part = 0.f;
        #pragma unroll
        for (int i = 0; i < 8; ++i) part += zs[i] * cent0[(size_t)k * D_ + lane + 32 * i];
        part = waveReduceSum(part);
        neg0 += __expf(part / dens0[k]);
    }
    float neg1 = 0.f;
    for (int k = 0; k < K1_; ++k) {
        float part = 0.f;
        #pragma unroll
        for (int i = 0; i < 8; ++i) part += zs[i] * cent1[(size_t)k * D_ + lane + 32 * i];
        part = waveReduceSum(part);
        neg1 += __expf(part / dens1[k]);
    }
    if (lane == 0) {
        neg0 = neg0 / (float)K0_ * (float)N_;
        neg1 = neg1 / (float)K1_ * (float)N_;
        const float l = (logf(neg0) - dot0 / dens0[k0]) +
                        (logf(neg1) - dot1 / dens1[k1]);
        atomicAdd(slot, l * (0.5f / (float)N_));
    }
}

__global__ void combine_loss(const float* __restrict__ slots, float* __restrict__ out) {
    if (threadIdx.x == 0 && blockIdx.x == 0)
        out[0] = LAM_ * slots[0] + (1.f - LAM_) * slots[1] + 0.3f * slots[2]
               + 0.01f * slots[3] + 0.01f * slots[4] + slots[5];
}

extern "C" void kernel_launch(void* const* d_in, const int* in_sizes, int n_in,
                              void* d_out, int out_size, void* d_ws, size_t ws_size,
                              hipStream_t stream) {
    (void)in_sizes; (void)n_in; (void)out_size; (void)ws_size;
    const float* z_mp  = (const float*)d_in[0];
    const float* z_sc  = (const float*)d_in[1];
    const float* feat  = (const float*)d_in[2];
    const float* pos   = (const float*)d_in[3];
    const int*   im2c0 = (const int*)d_in[4];
    const int*   im2c1 = (const int*)d_in[5];
    const float* cent0 = (const float*)d_in[6];
    const float* cent1 = (const float*)d_in[7];
    const float* dens0 = (const float*)d_in[8];
    const float* dens1 = (const float*)d_in[9];
    const float* pW1 = (const float*)d_in[10];
    const float* pb1 = (const float*)d_in[11];
    const float* pW2 = (const float*)d_in[12];
    const float* pb2 = (const float*)d_in[13];
    const float* mW1 = (const float*)d_in[14];
    const float* mb1 = (const float*)d_in[15];
    const float* mW2 = (const float*)d_in[16];
    const float* mb2 = (const float*)d_in[17];

    float* ws = (float*)d_ws;
    const size_t ND = (size_t)N_ * D_;
    float* projmp = ws;                 // z_proj_mp (== proj_mp_feat)
    float* projsc = projmp + ND;
    float* intra  = projsc + ND;
    float* projft = intra + ND;
    float* htmp   = projft + ND;
    float* nrm_mp = htmp + ND;
    float* nrm_sc = nrm_mp + N_;
    float* nrm_in = nrm_sc + N_;
    float* nrm_ft = nrm_in + N_;
    float* rsum   = nrm_ft + N_;                 // 5 x N
    float* rpos   = rsum + 5 * (size_t)N_;       // 5 x N
    float* S0     = rpos + 5 * (size_t)N_;       // K0 x D
    float* SP     = S0 + (size_t)K0_ * D_;       // K0*K1 x D
    float* n0     = SP + (size_t)K0_ * K1_ * D_; // K0
    float* npr    = n0 + K0_;                    // K0*K1
    float* slots  = npr + (size_t)K0_ * K1_;     // 8 loss slots

    const size_t zcount = (size_t)K0_ * D_ + (size_t)K0_ * K1_ * D_
                        + K0_ + (size_t)K0_ * K1_ + 8;
    hipMemsetAsync(S0, 0, zcount * sizeof(float), stream);

    const int gemmGrid = (N_ / 16) * (D_ / 16) / 4; // 1024 blocks x 4 waves
    // projection head (shared weights): mp, sc, feat
    gemm_xwt_bias<<<gemmGrid, 128, 0, stream>>>(z_mp, pW1, pb1, htmp, 1);
    gemm_xwt_bias<<<gemmGrid, 128, 0, stream>>>(htmp, pW2, pb2, projmp, 0);
    gemm_xwt_bias<<<gemmGrid, 128, 0, stream>>>(z_sc, pW1, pb1, htmp, 1);
    gemm_xwt_bias<<<gemmGrid, 128, 0, stream>>>(htmp, pW2, pb2, projsc, 0);
    gemm_xwt_bias<<<gemmGrid, 128, 0, stream>>>(feat, pW1, pb1, htmp, 1);
    gemm_xwt_bias<<<gemmGrid, 128, 0, stream>>>(htmp, pW2, pb2, projft, 0);
    // intra-view head
    gemm_xwt_bias<<<gemmGrid, 128, 0, stream>>>(z_mp, mW1, mb1, htmp, 1);
    gemm_xwt_bias<<<gemmGrid, 128, 0, stream>>>(htmp, mW2, mb2, intra, 0);

    row_norms<<<N_ / 8, 256, 0, stream>>>(projmp, nrm_mp);
    row_norms<<<N_ / 8, 256, 0, stream>>>(projsc, nrm_sc);
    row_norms<<<N_ / 8, 256, 0, stream>>>(intra,  nrm_in);
    row_norms<<<N_ / 8, 256, 0, stream>>>(projft, nrm_ft);

    const int simGrid = (N_ / 16) / 2; // 128 blocks x 2 waves
    // lori_mp: rows of m_mp2sc
    sim_nce_pass<<<simGrid, 64, 0, stream>>>(projmp, projsc, nrm_mp, nrm_sc, pos,
                                             1.f / TAU_, rsum + 0 * N_, rpos + 0 * N_);
    // lori_sc: rows of m_mp2sc^T == sim(sc, mp)
    sim_nce_pass<<<simGrid, 64, 0, stream>>>(projsc, projmp, nrm_sc, nrm_mp, pos,
                                             1.f / TAU_, rsum + 1 * N_, rpos + 1 * N_);
    // intra (tau = 0.05)
    sim_nce_pass<<<simGrid, 64, 0, stream>>>(intra, intra, nrm_in, nrm_in, pos,
                                             1.f / 0.05f, rsum + 2 * N_, rpos + 2 * N_);
    // lori_mpfeat / lori_featmp
    sim_nce_pass<<<simGrid, 64, 0, stream>>>(projmp, projft, nrm_mp, nrm_ft, pos,
                                             1.f / TAU_, rsum + 3 * N_, rpos + 3 * N_);
    sim_nce_pass<<<simGrid, 64, 0, stream>>>(projft, projmp, nrm_ft, nrm_mp, pos,
                                             1.f / TAU_, rsum + 4 * N_, rpos + 4 * N_);

    for (int p = 0; p < 5; ++p)
        nce_finalize<<<1, 256, 0, stream>>>(rsum + (size_t)p * N_,
                                            rpos + (size_t)p * N_, slots + p);

    pocl_scatter<<<(N_ * D_) / 256, 256, 0, stream>>>(z_mp, im2c0, im2c1, S0, SP, n0, npr);
    pocl_loss<<<N_ / 8, 256, 0, stream>>>(z_sc, im2c0, im2c1, S0, SP, n0, npr,
                                          cent0, cent1, dens0, dens1, slots + 5);
    combine_loss<<<1, 1, 0, stream>>>(slots, (float*)d_out);
}